// mamba_fusion1_26371099198481
// MI455X (gfx1250) — compile-verified
//
#include <hip/hip_runtime.h>
#include <hip/hip_bf16.h>
#include <math.h>

// ---------------------------------------------------------------------------
// MI455X (gfx1250) implementation of the VMamba-style fusion block.
// GEMMs run on the CDNA5 matrix pipe via V_WMMA_F32_16X16X4_F32 (full f32
// precision). LDS tiles are [m][k]/[n][k] so each fragment is one aligned
// ds_load_b64, and global->LDS staging uses gfx1250 async-load-to-LDS
// (ASYNCcnt + s_wait_asynccnt) with a ds-store fallback.
// Cross-scan permutations are folded into the scan kernel's index math; the
// 4-direction merge uses global f32 atomics.
// ---------------------------------------------------------------------------

typedef __attribute__((ext_vector_type(2))) float v2f;
typedef __attribute__((ext_vector_type(8))) float v8f;

#define TOK 2048   // B*H*W tokens
#define HWL 1024   // H*W
#define DIN 1536   // d_inner
#define CCH 768    // C

__device__ __forceinline__ float siluf(float x)     { return x / (1.f + __expf(-x)); }
__device__ __forceinline__ float softplusf(float x) { return (x > 20.f) ? x : log1pf(__expf(x)); }

// ---- gfx1250 async global->LDS copies (guarded: fall back to ds stores) ----
// Probe-learned signatures (clang prints AS1 as "__device__", AS3 as "__shared__"):
//   __builtin_amdgcn_global_load_async_to_lds_b128(int4 AS1*, int4 AS3*, Imm, Imm)
//   __builtin_amdgcn_global_load_async_to_lds_b32 (int  AS1*, int  AS3*, Imm, Imm)
#if defined(__HIP_DEVICE_COMPILE__) && \
    __has_builtin(__builtin_amdgcn_global_load_async_to_lds_b128) && \
    __has_builtin(__builtin_amdgcn_global_load_async_to_lds_b32)  && \
    __has_builtin(__builtin_amdgcn_s_wait_asynccnt)
#define HAVE_ASYNC_LDS 1
typedef int vi4 __attribute__((vector_size(16)));
__device__ __forceinline__ void async_cp_b128(const float* g, float* l) {
  __builtin_amdgcn_global_load_async_to_lds_b128(
      (__attribute__((address_space(1))) vi4*)g,
      (__attribute__((address_space(3))) vi4*)l, 0, 0);
}
__device__ __forceinline__ void async_cp_b32(const float* g, float* l) {
  __builtin_amdgcn_global_load_async_to_lds_b32(
      (__attribute__((address_space(1))) int*)g,
      (__attribute__((address_space(3))) int*)l, 0, 0);
}
__device__ __forceinline__ void async_wait0() { __builtin_amdgcn_s_wait_asynccnt(0); }
#else
#define HAVE_ASYNC_LDS 0
#endif

// ---------------- LayerNorm over BCHW input; also writes token-major raw copy
__global__ __launch_bounds__(256) void ln_bchw(const float* __restrict__ x,
                                               const float* __restrict__ w,
                                               const float* __restrict__ b,
                                               float* __restrict__ nOut,
                                               float* __restrict__ rawT)
{
  __shared__ float red[256];
  const int t  = blockIdx.x;
  const int bb = t >> 10, hw = t & 1023;
  const float* xp = x + (long)bb * CCH * HWL + hw;
  float s = 0.f, s2 = 0.f;
  for (int c = threadIdx.x; c < CCH; c += 256) {
    float v = xp[(long)c * HWL];
    rawT[(long)t * CCH + c] = v;
    s += v; s2 += v * v;
  }
  red[threadIdx.x] = s; __syncthreads();
  for (int o = 128; o > 0; o >>= 1) { if (threadIdx.x < o) red[threadIdx.x] += red[threadIdx.x + o]; __syncthreads(); }
  float mu = red[0] / CCH; __syncthreads();
  red[threadIdx.x] = s2; __syncthreads();
  for (int o = 128; o > 0; o >>= 1) { if (threadIdx.x < o) red[threadIdx.x] += red[threadIdx.x + o]; __syncthreads(); }
  float var  = red[0] / CCH - mu * mu;
  float rstd = rsqrtf(var + 1e-5f);
  for (int c = threadIdx.x; c < CCH; c += 256) {
    float v = rawT[(long)t * CCH + c];
    nOut[(long)t * CCH + c] = (v - mu) * rstd * w[c] + b[c];
  }
}

// ---------------- LayerNorm over token-major rows (optionally concat of two)
__global__ __launch_bounds__(256) void ln_rows(const float* __restrict__ in1,
                                               const float* __restrict__ in2,
                                               int C1, int Ct,
                                               const float* __restrict__ w,
                                               const float* __restrict__ b,
                                               float* __restrict__ out)
{
  __shared__ float red[256];
  const int t  = blockIdx.x;
  const int C2 = Ct - C1;
  float s = 0.f, s2 = 0.f;
  for (int c = threadIdx.x; c < Ct; c += 256) {
    float v = (c < C1) ? in1[(long)t * C1 + c] : in2[(long)t * C2 + (c - C1)];
    s += v; s2 += v * v;
  }
  red[threadIdx.x] = s; __syncthreads();
  for (int o = 128; o > 0; o >>= 1) { if (threadIdx.x < o) red[threadIdx.x] += red[threadIdx.x + o]; __syncthreads(); }
  float mu = red[0] / Ct; __syncthreads();
  red[threadIdx.x] = s2; __syncthreads();
  for (int o = 128; o > 0; o >>= 1) { if (threadIdx.x < o) red[threadIdx.x] += red[threadIdx.x + o]; __syncthreads(); }
  float var  = red[0] / Ct - mu * mu;
  float rstd = rsqrtf(var + 1e-5f);
  for (int c = threadIdx.x; c < Ct; c += 256) {
    float v = (c < C1) ? in1[(long)t * C1 + c] : in2[(long)t * C2 + (c - C1)];
    out[(long)t * Ct + c] = (v - mu) * rstd * w[c] + b[c];
  }
}

// ---------------- WMMA f32 GEMM: C = act(A @ W (+bias)) (+R0)(+R1)
// A: [M,K] row-major (lda), per-z element offset aZ (column shift for dt_proj).
// W: WT ? [N,K] : [K,N] (ldw), per-z element offset wZ.
// C row remap: row(m) = (m>>10)*cOuter + (m&1023); per-z pointer offset cZ.
// Block tile 64x128, 8 wave32; each wave owns a 32x32 tile (4 accumulators).
// LDS tiles are [m][k]/[n][k], row stride 20 floats (80B: 16B-aligned rows,
// conflict-free stride mod 64 banks) -> each fragment = one ds_load_b64.
template <bool WT, int ACT, bool HAS_BIAS, int NRES>
__global__ __launch_bounds__(256, 2) void gemm_wmma(
    const float* __restrict__ A, int lda, long aZ,
    const float* __restrict__ W, int ldw, long wZ,
    const float* __restrict__ bias, long biasZ,
    const float* __restrict__ R0, const float* __restrict__ R1, int ldr,
    float* __restrict__ Cc, int ldc, long cZ, int cOuter,
    int M, int N, int K)
{
  __shared__ float sA[64][20];    // [m][k]
  __shared__ float sB[128][20];   // [n][k]

  const int tid  = threadIdx.x;
  const int lane = tid & 31;
  const int wv   = tid >> 5;
  const int m0   = blockIdx.x * 64;
  const int n0   = blockIdx.y * 128;
  const int bz   = blockIdx.z;

  A += (long)bz * aZ;
  W += (long)bz * wZ;
  if (HAS_BIAS) bias += (long)bz * biasZ;
  float* Cp = Cc + (long)bz * cZ;

  v8f acc00 = {0.f,0.f,0.f,0.f,0.f,0.f,0.f,0.f};
  v8f acc01 = acc00, acc10 = acc00, acc11 = acc00;

  const int hf = lane >> 4;           // K half selector for A/B frags
  const int q  = lane & 15;
  const int fr = (wv >> 2) * 32;      // wave row base within 64
  const int fc = (wv & 3) * 32;       // wave col base within 128

  const int ar = tid >> 2;            // 0..63 : A row / W(T) row
  const int ak = (tid & 3) * 4;       // k quad

  for (int k0 = 0; k0 < K; k0 += 16) {
    // ---- stage A tile [m][k]
    {
      const float* ga = A + (long)(m0 + ar) * lda + (k0 + ak);
#if HAVE_ASYNC_LDS
      async_cp_b128(ga, &sA[ar][ak]);
#else
      const float4 v = *(const float4*)ga;
      *(float4*)&sA[ar][ak] = v;
#endif
    }
    // ---- stage B tile [n][k]
    if (WT) {
#pragma unroll
      for (int h2 = 0; h2 < 2; ++h2) {
        const int n = n0 + ar + 64 * h2;
        if (n < N) {
#if HAVE_ASYNC_LDS
          async_cp_b128(W + (long)n * ldw + (k0 + ak), &sB[ar + 64 * h2][ak]);
#else
          const float4 v = *(const float4*)(W + (long)n * ldw + (k0 + ak));
          *(float4*)&sB[ar + 64 * h2][ak] = v;
#endif
        } else {
          const float4 z = {0.f, 0.f, 0.f, 0.f};
          *(float4*)&sB[ar + 64 * h2][ak] = z;
        }
      }
    } else {
      const int kk = tid >> 4;         // 0..15
      const int nq = (tid & 15) * 8;   // 0..120
#pragma unroll
      for (int j = 0; j < 8; ++j) {
        const int n = n0 + nq + j;
        if (n < N) {
#if HAVE_ASYNC_LDS
          async_cp_b32(W + (long)(k0 + kk) * ldw + n, &sB[nq + j][kk]);
#else
          sB[nq + j][kk] = W[(long)(k0 + kk) * ldw + n];
#endif
        } else {
          sB[nq + j][kk] = 0.f;
        }
      }
    }
    // prefetch next K tile while we compute this one
    if (k0 + 16 < K) {
      __builtin_prefetch(A + (long)(m0 + ar) * lda + (k0 + 16 + ak), 0, 1);
      if (WT) { if (n0 + ar < N) __builtin_prefetch(W + (long)(n0 + ar) * ldw + (k0 + 16 + ak), 0, 1); }
      else    { __builtin_prefetch(W + (long)(k0 + 16 + (tid >> 4)) * ldw + (n0 + (tid & 15) * 8), 0, 1); }
    }
#if HAVE_ASYNC_LDS
    async_wait0();
#endif
    __syncthreads();

#pragma unroll
    for (int kk = 0; kk < 16; kk += 4) {
      // A 16x4 frag: lanes 0-15 hold K{kk,kk+1}, lanes 16-31 K{kk+2,kk+3}
      const v2f a0 = *(const v2f*)&sA[fr + q][kk + 2 * hf];
      const v2f a1 = *(const v2f*)&sA[fr + 16 + q][kk + 2 * hf];
      // B 4x16 frag: vgpr pair rows {2hf, 2hf+1}
      const v2f b0 = *(const v2f*)&sB[fc + q][kk + 2 * hf];
      const v2f b1 = *(const v2f*)&sB[fc + 16 + q][kk + 2 * hf];
      acc00 = __builtin_amdgcn_wmma_f32_16x16x4_f32(false, a0, false, b0, (short)0, acc00, false, false);
      acc01 = __builtin_amdgcn_wmma_f32_16x16x4_f32(false, a0, false, b1, (short)0, acc01, false, false);
      acc10 = __builtin_amdgcn_wmma_f32_16x16x4_f32(false, a1, false, b0, (short)0, acc10, false, false);
      acc11 = __builtin_amdgcn_wmma_f32_16x16x4_f32(false, a1, false, b1, (short)0, acc11, false, false);
    }
    __syncthreads();
  }

  // ---- epilogue: D layout vgpr i -> M=i (lanes 0-15) / M=i+8 (lanes 16-31)
  auto emit = [&](float v, int m, int n) {
    if (n >= N) return;
    if (HAS_BIAS) v += bias[n];
    if (ACT == 1) v = softplusf(v);
    if (NRES >= 1) v += R0[(long)m * ldr + n];
    if (NRES >= 2) v += R1[(long)m * ldr + n];
    const long rm = (long)(m >> 10) * cOuter + (m & 1023);
    Cp[rm * ldc + n] = v;
  };
#pragma unroll
  for (int i = 0; i < 8; ++i) {
    const int mA = m0 + fr + i + 8 * hf;
    const int mB = mA + 16;
    const int nA = n0 + fc + q;
    const int nB = nA + 16;
    emit(acc00[i], mA, nA);
    emit(acc01[i], mA, nB);
    emit(acc10[i], mB, nA);
    emit(acc11[i], mB, nB);
  }
}

// ---------------- depthwise 3x3 conv (SAME, zero pad) + bias + SiLU
__global__ __launch_bounds__(256) void conv3x3_silu(const float* __restrict__ X,
                                                    const float* __restrict__ cw,
                                                    const float* __restrict__ cb,
                                                    float* __restrict__ xc)
{
  const int d  = blockIdx.x * blockDim.x + threadIdx.x;  // 0..1535
  const int t  = blockIdx.y;                             // 0..2047
  const int bb = t >> 10, hw = t & 1023;
  const int h  = hw >> 5, wq = hw & 31;
  float acc = cb[d];
#pragma unroll
  for (int kh = 0; kh < 3; ++kh) {
    const int hh = h + kh - 1;
    if (hh < 0 || hh > 31) continue;
#pragma unroll
    for (int kw = 0; kw < 3; ++kw) {
      const int ww = wq + kw - 1;
      if (ww < 0 || ww > 31) continue;
      acc += X[((long)(bb * HWL + hh * 32 + ww)) * DIN + d] * cw[d * 9 + kh * 3 + kw];
    }
  }
  xc[(long)t * DIN + d] = siluf(acc);
}

// ---------------- selective scan: one thread per (b,k,d) channel, n=4 state.
__global__ __launch_bounds__(256) void scan_kernel(const float* __restrict__ xc,    // (TOK, DIN)
                                                   const float* __restrict__ delta, // ((b*4+k)*HWL+pos)*DIN+d
                                                   const float* __restrict__ xdbl,  // (t)*224 + k*56 + {48..55}
                                                   const float* __restrict__ Alog,  // (k*DIN+d)*4+n
                                                   const float* __restrict__ Dp,    // k*DIN+d
                                                   float* __restrict__ ymerge)      // (TOK, DIN) accumulated
{
  const int g = blockIdx.x * blockDim.x + threadIdx.x;
  if (g >= 2 * 4 * DIN) return;
  const int d = g % DIN;
  const int k = (g / DIN) & 3;
  const int b = g / (4 * DIN);

  const float A0 = -__expf(Alog[((k * DIN + d) << 2) + 0]);
  const float A1 = -__expf(Alog[((k * DIN + d) << 2) + 1]);
  const float A2 = -__expf(Alog[((k * DIN + d) << 2) + 2]);
  const float A3 = -__expf(Alog[((k * DIN + d) << 2) + 3]);
  const float Dv = Dp[k * DIN + d];

  float h0 = 0.f, h1 = 0.f, h2 = 0.f, h3 = 0.f;
  for (int l = 0; l < HWL; ++l) {
    int pos;
    if      (k == 0) pos = l;
    else if (k == 1) pos = ((l & 31) << 5) | (l >> 5);
    else if (k == 2) pos = 1023 - l;
    else { const int r = 1023 - l; pos = ((r & 31) << 5) | (r >> 5); }
    const int t = b * HWL + pos;

    const float u  = xc[(long)t * DIN + d];
    const float dt = delta[((long)((b << 2) + k) * HWL + pos) * DIN + d];
    const float* xr = xdbl + (long)t * 224 + k * 56;
    const float B0 = xr[48], B1 = xr[49], B2 = xr[50], B3 = xr[51];
    const float C0 = xr[52], C1 = xr[53], C2 = xr[54], C3 = xr[55];

    const float du = dt * u;
    h0 = h0 * __expf(dt * A0) + du * B0;
    h1 = h1 * __expf(dt * A1) + du * B1;
    h2 = h2 * __expf(dt * A2) + du * B2;
    h3 = h3 * __expf(dt * A3) + du * B3;

    const float y = h0 * C0 + h1 * C1 + h2 * C2 + h3 * C3 + u * Dv;
    atomicAdd(&ymerge[(long)t * DIN + d], y);
  }
}

// ---------------- small elementwise helpers
__global__ __launch_bounds__(256) void zero_f32(float* __restrict__ p, long n)
{
  const long i = (long)blockIdx.x * 256 + threadIdx.x;
  if (i < n) p[i] = 0.f;
}

__global__ __launch_bounds__(256) void mul_silu(const float* __restrict__ a,
                                                const float* __restrict__ g,
                                                float* __restrict__ out, long n)
{
  const long i = (long)blockIdx.x * 256 + threadIdx.x;
  if (i < n) out[i] = a[i] * siluf(g[i]);
}

__global__ __launch_bounds__(256) void to_bchw(const float* __restrict__ in,
                                               float* __restrict__ out)
{
  const long i = (long)blockIdx.x * 256 + threadIdx.x;  // over B*C*H*W
  if (i >= (long)2 * CCH * HWL) return;
  const int hw = (int)(i & 1023);
  const int c  = (int)((i >> 10) % CCH);
  const int bb = (int)(i / ((long)CCH * HWL));
  out[i] = in[((long)bb * HWL + hw) * CCH + c];
}

// ---------------------------------------------------------------------------
// Host orchestration
// ---------------------------------------------------------------------------
struct SsmP {
  const float *xw, *dtw, *dtb, *Alog, *Dv, *cw, *cb, *onw, *onb;
};

static SsmP load_ssm(void* const* d, int base)
{
  SsmP p;
  p.xw   = (const float*)d[base + 0];
  p.dtw  = (const float*)d[base + 1];
  p.dtb  = (const float*)d[base + 2];
  p.Alog = (const float*)d[base + 3];
  p.Dv   = (const float*)d[base + 4];
  p.cw   = (const float*)d[base + 5];
  p.cb   = (const float*)d[base + 6];
  p.onw  = (const float*)d[base + 7];
  p.onb  = (const float*)d[base + 8];
  return p;
}

static inline unsigned gy(int N) { return (unsigned)((N + 127) / 128); }

static void run_ss2d(const float* X, const SsmP& P,
                     float* xc, float* xdbl, float* delta, float* ymerge, float* yout,
                     hipStream_t s)
{
  // depthwise conv + SiLU
  conv3x3_silu<<<dim3(DIN / 256, TOK), 256, 0, s>>>(X, P.cw, P.cb, xc);
  // x_proj for all 4 directions at once: N = 4*56 = 224, W layout [N,K]
  gemm_wmma<true, 0, false, 0><<<dim3(TOK / 64, gy(224), 1), 256, 0, s>>>(
      xc, DIN, 0, P.xw, DIN, 0, nullptr, 0, nullptr, nullptr, 0,
      xdbl, 224, 0, 1024, TOK, 224, DIN);
  // delta = softplus(dts @ dt_w^T + dt_b), batched over direction k (grid.z)
  gemm_wmma<true, 1, true, 0><<<dim3(TOK / 64, gy(DIN), 4), 256, 0, s>>>(
      xdbl, 224, 56, P.dtw, 48, (long)DIN * 48, P.dtb, DIN, nullptr, nullptr, 0,
      delta, DIN, (long)HWL * DIN, 4096, TOK, DIN, 48);
  // scan + merge
  zero_f32<<<(TOK * DIN + 255) / 256, 256, 0, s>>>(ymerge, (long)TOK * DIN);
  scan_kernel<<<(2 * 4 * DIN + 255) / 256, 256, 0, s>>>(xc, delta, xdbl, P.Alog, P.Dv, ymerge);
  // out-norm
  ln_rows<<<TOK, 256, 0, s>>>(ymerge, nullptr, DIN, DIN, P.onw, P.onb, yout);
}

extern "C" void kernel_launch(void* const* d_in, const int* in_sizes, int n_in,
                              void* d_out, int out_size, void* d_ws, size_t ws_size,
                              hipStream_t stream)
{
  (void)in_sizes; (void)n_in; (void)out_size; (void)ws_size;

  // Inputs (setup_inputs dict insertion order, nested dicts depth-first):
  const float* x1  = (const float*)d_in[0];
  const float* x2  = (const float*)d_in[1];
  const float* cnw = (const float*)d_in[2];
  const float* cnb = (const float*)d_in[3];
  const float* ipr = (const float*)d_in[4];   // (768,1536)  [K,N]
  const float* ipe = (const float*)d_in[5];
  const float* opr = (const float*)d_in[6];   // (1536,768)  [K,N]
  const float* ope = (const float*)d_in[7];
  const SsmP  sr   = load_ssm(d_in, 8);
  const SsmP  se   = load_ssm(d_in, 17);
  const float* qnw = (const float*)d_in[26];
  const float* qnb = (const float*)d_in[27];
  const float* qin = (const float*)d_in[28];  // (1536,3072) [K,N]
  const float* qop = (const float*)d_in[29];  // (1536,768)  [K,N]
  const SsmP  sc   = load_ssm(d_in, 30);

  // Workspace layout (floats)
  float* wsf = (float*)d_ws;
  const long TD = (long)TOK * DIN, TC = (long)TOK * CCH;
  float* S0  = wsf;                // norm output (2048x1536); cross: n1|n2 halves
  float* S1  = S0  + TD;           // a / xm
  float* S2  = S1  + TD;           // c / z
  float* S3  = S2  + TD;           // xc
  float* S4  = S3  + TD;           // xdbl (2048x224), padded slot of 2048x256
  float* S5  = S4  + (long)TOK * 256;          // delta (2,4,1024,1536)
  float* S6  = S5  + (long)2 * 4 * HWL * DIN;  // ymerge / fusedT
  float* S7  = S6  + TD;           // ss2d output (post out-norm)
  float* S8  = S7  + TD;           // y1 / y
  float* S9  = S8  + TD;           // y2
  float* S10 = S9  + TD;           // x1 token-major
  float* S11 = S10 + TC;           // x2 token-major
  float* S12 = S11 + TC;           // x1'
  float* S13 = S12 + TC;           // x2'
  float* n1  = S0;
  float* n2  = S0 + TC;

  // ---- cross block
  ln_bchw<<<TOK, 256, 0, stream>>>(x1, cnw, cnb, n1, S10);
  ln_bchw<<<TOK, 256, 0, stream>>>(x2, cnw, cnb, n2, S11);

  gemm_wmma<false, 0, false, 0><<<dim3(TOK / 64, gy(DIN), 1), 256, 0, stream>>>(
      n1, CCH, 0, ipr, DIN, 0, nullptr, 0, nullptr, nullptr, 0,
      S1, DIN, 0, 1024, TOK, DIN, CCH);
  gemm_wmma<false, 0, false, 0><<<dim3(TOK / 64, gy(DIN), 1), 256, 0, stream>>>(
      n2, CCH, 0, ipe, DIN, 0, nullptr, 0, nullptr, nullptr, 0,
      S2, DIN, 0, 1024, TOK, DIN, CCH);

  run_ss2d(S1, sr, S3, S4, S5, S6, S7, stream);
  mul_silu<<<(TD + 255) / 256, 256, 0, stream>>>(S7, S2, S8, TD);   // y1
  run_ss2d(S2, se, S3, S4, S5, S6, S7, stream);
  mul_silu<<<(TD + 255) / 256, 256, 0, stream>>>(S7, S1, S9, TD);   // y2

  gemm_wmma<false, 0, false, 1><<<dim3(TOK / 64, gy(CCH), 1), 256, 0, stream>>>(
      S8, DIN, 0, opr, CCH, 0, nullptr, 0, S10, nullptr, CCH,
      S12, CCH, 0, 1024, TOK, CCH, DIN);   // x1' = x1 + y1 @ out_proj_r
  gemm_wmma<false, 0, false, 1><<<dim3(TOK / 64, gy(CCH), 1), 256, 0, stream>>>(
      S9, DIN, 0, ope, CCH, 0, nullptr, 0, S11, nullptr, CCH,
      S13, CCH, 0, 1024, TOK, CCH, DIN);   // x2' = x2 + y2 @ out_proj_e

  // ---- concat block
  ln_rows<<<TOK, 256, 0, stream>>>(S12, S13, CCH, DIN, qnw, qnb, S0);

  gemm_wmma<false, 0, false, 0><<<dim3(TOK / 64, gy(DIN), 1), 256, 0, stream>>>(
      S0, DIN, 0, qin, 2 * DIN, 0, nullptr, 0, nullptr, nullptr, 0,
      S1, DIN, 0, 1024, TOK, DIN, DIN);            // xm
  gemm_wmma<false, 0, false, 0><<<dim3(TOK / 64, gy(DIN), 1), 256, 0, stream>>>(
      S0, DIN, 0, qin + DIN, 2 * DIN, 0, nullptr, 0, nullptr, nullptr, 0,
      S2, DIN, 0, 1024, TOK, DIN, DIN);            // z

  run_ss2d(S1, sc, S3, S4, S5, S6, S7, stream);
  mul_silu<<<(TD + 255) / 256, 256, 0, stream>>>(S7, S2, S8, TD);   // y

  gemm_wmma<false, 0, false, 2><<<dim3(TOK / 64, gy(CCH), 1), 256, 0, stream>>>(
      S8, DIN, 0, qop, CCH, 0, nullptr, 0, S12, S13, CCH,
      S6, CCH, 0, 1024, TOK, CCH, DIN);   // fused = x1' + x2' + y @ out_proj

  to_bchw<<<(int)((2L * CCH * HWL + 255) / 256), 256, 0, stream>>>(S6, (float*)d_out);
}